// PointCloudFeaturePredictor_6047313953093
// MI455X (gfx1250) — compile-verified
//
#include <hip/hip_runtime.h>
#include <math.h>

#define N1C 65536
#define N2C 65536
#define CF  64
#define KNN 8

typedef float v2f __attribute__((ext_vector_type(2)));
typedef float v8f __attribute__((ext_vector_type(8)));

__device__ __forceinline__ v8f wmma4(v2f a, v2f b, v8f c) {
  // D = A(16x4,f32) * B(4x16,f32) + C(16x16,f32)
  return __builtin_amdgcn_wmma_f32_16x16x4_f32(false, a, false, b, (short)0, c, false, false);
}

__device__ __forceinline__ unsigned f2ord(float f) {
  unsigned u = __float_as_uint(f);
  return (u & 0x80000000u) ? ~u : (u | 0x80000000u);
}
__device__ __forceinline__ float ord2f(unsigned u) {
  return (u & 0x80000000u) ? __uint_as_float(u & 0x7fffffffu) : __uint_as_float(~u);
}

// ---------------- init: reset global max/sum accumulators ----------------
__global__ void init_kernel(unsigned* __restrict__ gmax, float* __restrict__ gsum) {
  int t = threadIdx.x;
  if (t < 8) { gmax[t] = 0u; gsum[t] = 0.0f; }
}

// ---------------- Kernel 1: brute-force KNN (top-8) ----------------
#define TS 256
__global__ void knn_kernel(const float* __restrict__ c1, const float* __restrict__ c2,
                           int* __restrict__ knn_idx) {
  __shared__ float tile[TS * 3];
  const int q = blockIdx.x * blockDim.x + threadIdx.x;
  const float qx = c2[q * 3 + 0], qy = c2[q * 3 + 1], qz = c2[q * 3 + 2];

  float best[KNN];
  int   bidx[KNN];
#pragma unroll
  for (int i = 0; i < KNN; ++i) { best[i] = 3.4e38f; bidx[i] = 0; }

  for (int base = 0; base < N1C; base += TS) {
    __syncthreads();
    for (int i = threadIdx.x; i < TS * 3; i += blockDim.x) tile[i] = c1[base * 3 + i];
    if (base + TS < N1C) __builtin_prefetch(&c1[(size_t)(base + TS) * 3], 0, 1);
    __syncthreads();
#pragma unroll 4
    for (int j = 0; j < TS; ++j) {
      const float dx = tile[j * 3 + 0] - qx;
      const float dy = tile[j * 3 + 1] - qy;
      const float dz = tile[j * 3 + 2] - qz;
      float d2 = dx * dx + dy * dy + dz * dz;
      if (d2 < best[KNN - 1]) {
        float d = d2; int id = base + j;
#pragma unroll
        for (int s = 0; s < KNN; ++s) {
          if (d < best[s]) {
            float tf = best[s]; int ti = bidx[s];
            best[s] = d; bidx[s] = id; d = tf; id = ti;
          }
        }
      }
    }
  }
#pragma unroll
  for (int i = 0; i < KNN; ++i) knn_idx[q * KNN + i] = bidx[i];
}

// ---------------- Kernel 2: build fea_agg rows + weight MLP (68->128->64->1) ----------------
// One wave = 16 rows (2 queries x 8 neighbors). Block = 64 threads = 2 waves.
__global__ void wmlp_kernel(const float* __restrict__ c1, const float* __restrict__ c2,
                            const float* __restrict__ f1, const int* __restrict__ knn_idx,
                            const float* __restrict__ W1, const float* __restrict__ b1,
                            const float* __restrict__ W2, const float* __restrict__ b2,
                            const float* __restrict__ W3, const float* __restrict__ b3,
                            float* __restrict__ wlogit) {
  __shared__ float As[2][16 * 68];
  __shared__ float H1s[2][16 * 128];
  __shared__ float H2s[2][16 * 64];
  __shared__ float IVs[2][16];

  const int wave = threadIdx.x >> 5;
  const int lane = threadIdx.x & 31;
  const int rbase = (blockIdx.x * 2 + wave) * 16;   // global row = q*8 + k

  float* A  = As[wave];
  float* h1 = H1s[wave];
  float* h2 = H2s[wave];
  float* iv = IVs[wave];

  // --- build grouped_xyz + 1/dist ---
  if (lane < 16) {
    const int r  = rbase + lane;
    const int q  = r >> 3;
    const int nb = knn_idx[r];
    const float dx = c1[nb * 3 + 0] - c2[q * 3 + 0];
    const float dy = c1[nb * 3 + 1] - c2[q * 3 + 1];
    const float dz = c1[nb * 3 + 2] - c2[q * 3 + 2];
    float dist = sqrtf(dx * dx + dy * dy + dz * dz);
    dist = fmaxf(dist, 1e-10f);
    A[lane * 68 + 0] = dx;
    A[lane * 68 + 1] = dy;
    A[lane * 68 + 2] = dz;
    iv[lane] = 1.0f / dist;
  }
  __syncthreads();
  // --- coords_weight = (1/d) / sum_k (1/d) ---
  if (lane < 16) {
    const int kq = lane & ~7;
    float norm = 0.0f;
#pragma unroll
    for (int kk = 0; kk < 8; ++kk) norm += iv[kq + kk];
    A[lane * 68 + 67] = iv[lane] / norm;
  }
  // --- gather neighbor feats (cols 3..66) ---
  for (int r = 0; r < 16; ++r) {
    const int nb = knn_idx[rbase + r];
    A[r * 68 + 3 + lane]      = f1[(size_t)nb * CF + lane];
    A[r * 68 + 3 + 32 + lane] = f1[(size_t)nb * CF + 32 + lane];
  }
  __syncthreads();

  const int m  = lane & 15;
  const int hi = lane >> 4;

  // --- GEMM1: (16x68) @ W1(68x128), bias+relu -> h1 ---
  for (int nt = 0; nt < 8; ++nt) {
    v8f c = {0.f, 0.f, 0.f, 0.f, 0.f, 0.f, 0.f, 0.f};
    const int col = nt * 16 + m;
#pragma unroll
    for (int ks = 0; ks < 17; ++ks) {
      const int k0 = ks * 4 + 2 * hi;
      v2f a, b;
      a.x = A[m * 68 + k0];
      a.y = A[m * 68 + k0 + 1];
      b.x = W1[k0 * 128 + col];
      b.y = W1[(k0 + 1) * 128 + col];
      c = wmma4(a, b, c);
    }
    const float bias = b1[col];
#pragma unroll
    for (int v = 0; v < 8; ++v) {
      h1[(v + 8 * hi) * 128 + col] = fmaxf(c[v] + bias, 0.0f);
    }
  }
  __syncthreads();

  // --- GEMM2: (16x128) @ W2(128x64), bias+relu -> h2 ---
  for (int nt = 0; nt < 4; ++nt) {
    v8f c = {0.f, 0.f, 0.f, 0.f, 0.f, 0.f, 0.f, 0.f};
    const int col = nt * 16 + m;
#pragma unroll
    for (int ks = 0; ks < 32; ++ks) {
      const int k0 = ks * 4 + 2 * hi;
      v2f a, b;
      a.x = h1[m * 128 + k0];
      a.y = h1[m * 128 + k0 + 1];
      b.x = W2[k0 * 64 + col];
      b.y = W2[(k0 + 1) * 64 + col];
      c = wmma4(a, b, c);
    }
    const float bias = b2[col];
#pragma unroll
    for (int v = 0; v < 8; ++v) {
      h2[(v + 8 * hi) * 64 + col] = fmaxf(c[v] + bias, 0.0f);
    }
  }
  __syncthreads();

  // --- GEMV: h2(16x64) @ W3(64x1) + b3 -> logits ---
  if (lane < 16) {
    float acc = b3[0];
#pragma unroll 8
    for (int n = 0; n < 64; ++n) acc += h2[lane * 64 + n] * W3[n];
    wlogit[rbase + lane] = acc;
  }
}

// ---------------- Kernel 3: column max over the N2 axis (per k) ----------------
__global__ void colmax_kernel(const float* __restrict__ wl, unsigned* __restrict__ gmax) {
  __shared__ unsigned smax[8];
  const int t = threadIdx.x;
  if (t < 8) smax[t] = 0u;
  __syncthreads();
  const int i = blockIdx.x * blockDim.x + t;
  atomicMax(&smax[i & 7], f2ord(wl[i]));
  __syncthreads();
  if (t < 8) atomicMax(&gmax[t], smax[t]);
}

// ---------------- Kernel 4: exp(w - max), column sums ----------------
__global__ void expsum_kernel(float* __restrict__ wl, const unsigned* __restrict__ gmax,
                              float* __restrict__ gsum) {
  __shared__ float ssum[8];
  const int t = threadIdx.x;
  if (t < 8) ssum[t] = 0.0f;
  __syncthreads();
  const int i = blockIdx.x * blockDim.x + t;
  const float mx = ord2f(gmax[i & 7]);
  const float e = __expf(wl[i] - mx);
  wl[i] = e;
  atomicAdd(&ssum[i & 7], e);
  __syncthreads();
  if (t < 8) atomicAdd(&gsum[t], ssum[t]);
}

// ---------------- Kernel 5: interpolated = sum_k softmax(w) * neighbor_feats ----------------
__global__ void interp_kernel(const float* __restrict__ f1, const int* __restrict__ knn_idx,
                              const float* __restrict__ wl, const float* __restrict__ gsum,
                              float* __restrict__ interp) {
  const int t  = blockIdx.x * blockDim.x + threadIdx.x;
  const int q  = t >> 6;
  const int ch = t & 63;
  float acc = 0.0f;
#pragma unroll
  for (int k = 0; k < KNN; ++k) {
    const float w = wl[q * KNN + k] / gsum[k];
    acc += w * f1[(size_t)knn_idx[q * KNN + k] * CF + ch];
  }
  interp[t] = acc;
}

// ---------------- Kernel 6: output MLP 64->256->64, predicted + residual ----------------
// One wave = 16 queries. Block = 64 threads = 2 waves.
__global__ void outmlp_kernel(const float* __restrict__ interp, const float* __restrict__ f2,
                              const float* __restrict__ Wo1, const float* __restrict__ bo1,
                              const float* __restrict__ Wo2, const float* __restrict__ bo2,
                              float* __restrict__ out) {
  __shared__ float Ts_[2][16 * 256];

  const int wave = threadIdx.x >> 5;
  const int lane = threadIdx.x & 31;
  const int qbase = (blockIdx.x * 2 + wave) * 16;
  float* T = Ts_[wave];

  const int m  = lane & 15;
  const int hi = lane >> 4;

  // --- GEMM1: interp(16x64) @ Wo1(64x256), bias+relu -> T ---
  for (int nt = 0; nt < 16; ++nt) {
    v8f c = {0.f, 0.f, 0.f, 0.f, 0.f, 0.f, 0.f, 0.f};
    const int col = nt * 16 + m;
#pragma unroll
    for (int ks = 0; ks < 16; ++ks) {
      const int k0 = ks * 4 + 2 * hi;
      v2f a, b;
      a.x = interp[(size_t)(qbase + m) * CF + k0];
      a.y = interp[(size_t)(qbase + m) * CF + k0 + 1];
      b.x = Wo1[k0 * 256 + col];
      b.y = Wo1[(k0 + 1) * 256 + col];
      c = wmma4(a, b, c);
    }
    const float bias = bo1[col];
#pragma unroll
    for (int v = 0; v < 8; ++v) {
      T[(v + 8 * hi) * 256 + col] = fmaxf(c[v] + bias, 0.0f);
    }
  }
  __syncthreads();

  // --- GEMM2: T(16x256) @ Wo2(256x64) + bo2 -> predicted; residual = feats2 - predicted ---
  for (int nt = 0; nt < 4; ++nt) {
    v8f c = {0.f, 0.f, 0.f, 0.f, 0.f, 0.f, 0.f, 0.f};
    const int col = nt * 16 + m;
#pragma unroll
    for (int ks = 0; ks < 64; ++ks) {
      const int k0 = ks * 4 + 2 * hi;
      v2f a, b;
      a.x = T[m * 256 + k0];
      a.y = T[m * 256 + k0 + 1];
      b.x = Wo2[k0 * 64 + col];
      b.y = Wo2[(k0 + 1) * 64 + col];
      c = wmma4(a, b, c);
    }
    const float bias = bo2[col];
#pragma unroll
    for (int v = 0; v < 8; ++v) {
      const int q = qbase + v + 8 * hi;
      const float p = c[v] + bias;
      out[(size_t)q * CF + col] = p;                                  // predicted
      out[(size_t)N2C * CF + (size_t)q * CF + col] = f2[(size_t)q * CF + col] - p;  // residual
    }
  }
}

extern "C" void kernel_launch(void* const* d_in, const int* in_sizes, int n_in,
                              void* d_out, int out_size, void* d_ws, size_t ws_size,
                              hipStream_t stream) {
  (void)in_sizes; (void)n_in; (void)out_size; (void)ws_size;

  const float* c1  = (const float*)d_in[0];
  const float* c2  = (const float*)d_in[1];
  const float* f1  = (const float*)d_in[2];
  const float* f2  = (const float*)d_in[3];
  const float* W1  = (const float*)d_in[4];
  const float* b1  = (const float*)d_in[5];
  const float* W2  = (const float*)d_in[6];
  const float* b2  = (const float*)d_in[7];
  const float* W3  = (const float*)d_in[8];
  const float* b3  = (const float*)d_in[9];
  const float* Wo1 = (const float*)d_in[10];
  const float* bo1 = (const float*)d_in[11];
  const float* Wo2 = (const float*)d_in[12];
  const float* bo2 = (const float*)d_in[13];
  float* out = (float*)d_out;

  char* ws = (char*)d_ws;
  int*      knn_idx = (int*)ws;                                          //  2 MB
  float*    wl      = (float*)(ws + (size_t)N2C * KNN * 4);              //  2 MB
  float*    itp     = (float*)(ws + (size_t)N2C * KNN * 8);              // 16 MB
  unsigned* gmax    = (unsigned*)(ws + (size_t)N2C * KNN * 8 + (size_t)N2C * CF * 4);
  float*    gsum    = (float*)((char*)gmax + 64);

  init_kernel<<<1, 32, 0, stream>>>(gmax, gsum);
  knn_kernel<<<N2C / 256, 256, 0, stream>>>(c1, c2, knn_idx);
  wmlp_kernel<<<(N2C * KNN) / 32, 64, 0, stream>>>(c1, c2, f1, knn_idx,
                                                   W1, b1, W2, b2, W3, b3, wl);
  colmax_kernel<<<(N2C * KNN) / 256, 256, 0, stream>>>(wl, gmax);
  expsum_kernel<<<(N2C * KNN) / 256, 256, 0, stream>>>(wl, gmax, gsum);
  interp_kernel<<<(N2C * CF) / 256, 256, 0, stream>>>(f1, knn_idx, wl, gsum, itp);
  outmlp_kernel<<<N2C / 32, 64, 0, stream>>>(itp, f2, Wo1, bo1, Wo2, bo2, out);
}